// SkipGram_57423712747539
// MI455X (gfx1250) — compile-verified
//
#include <hip/hip_runtime.h>
#include <hip/hip_bf16.h>

typedef __attribute__((ext_vector_type(2))) float v2f;
typedef __attribute__((ext_vector_type(8))) float v8f;

#define EMB   128
#define NEGS  10
#define BATCH 65536
#define TILE  16                 // batch elements per wave (one 16x16 WMMA tile)
#define WPB   8                  // waves per block
#define BLOCK (WPB * 32)
#define GRID  (BATCH / (TILE * WPB))   // 512 blocks -> 4096 waves

__device__ __forceinline__ float log_sigmoid(float x) {
    // log(sigmoid(x)) = min(x,0) - log1p(exp(-|x|))   (numerically stable)
    return fminf(x, 0.0f) - log1pf(__expf(-fabsf(x)));
}

__global__ void skipgram_wmma_kernel(const float* __restrict__ W_in,
                                     const float* __restrict__ W_ctx,
                                     const int*  __restrict__ input_word,
                                     const int*  __restrict__ context_word,
                                     const int*  __restrict__ neg,
                                     float* __restrict__ partials) {
    __shared__ float wsum[WPB];
    const int tid  = threadIdx.x;
    const int lane = tid & 31;
    const int wave = tid >> 5;

    const int tile = blockIdx.x * WPB + wave;
    const int n = lane & 15;   // batch element within tile == A row (M) == B col (N)
    const int h = lane >> 4;   // lane half: selects K sub-pair within a 4-wide K step
    const int b = tile * TILE + n;

    // ---- A fragments (emb_input rows), all 32 K-steps, held in registers ----
    // f32 16x4 A layout (documented): lane L row M=L%16; VGPR v holds K = 4s + 2h + v
    const float* arow = W_in + (size_t)input_word[b] * EMB;
    v2f aFrag[32];
#pragma unroll
    for (int s = 0; s < 32; ++s)
        aFrag[s] = *reinterpret_cast<const v2f*>(arow + 4 * s + 2 * h);

    const bool holder = (lane < 8) || (lane >= 24);  // lanes owning a diagonal elem
    const int  r      = lane & 7;                    // acc VGPR index of that elem
    float lsum = 0.0f;

    for (int j = 0; j <= NEGS; ++j) {
        const int idx = (j == 0) ? context_word[b] : neg[b * NEGS + (j - 1)];
        const float* brow = W_ctx + (size_t)idx * EMB;
        v8f acc = {0.f, 0.f, 0.f, 0.f, 0.f, 0.f, 0.f, 0.f};
#pragma unroll
        for (int s = 0; s < 32; ++s) {
            v2f bf = *reinterpret_cast<const v2f*>(brow + 4 * s + 2 * h);
            // D = A(16x4) * B(4x16) + C, f32 throughout
            acc = __builtin_amdgcn_wmma_f32_16x16x4_f32(
                false, aFrag[s], false, bf, (short)0, acc, false, false);
        }
        // diag element (d,d): lane (d<8 ? d : d+16), accumulator VGPR d&7
        float d = acc[0];
        d = (r == 1) ? acc[1] : d;
        d = (r == 2) ? acc[2] : d;
        d = (r == 3) ? acc[3] : d;
        d = (r == 4) ? acc[4] : d;
        d = (r == 5) ? acc[5] : d;
        d = (r == 6) ? acc[6] : d;
        d = (r == 7) ? acc[7] : d;
        const float x = (j == 0) ? d : -d;       // neg_score = -dot
        const float c = log_sigmoid(x);
        lsum += holder ? c : 0.0f;
    }

    // deterministic reduction: wave shfl-xor -> LDS slots -> per-block partial
#pragma unroll
    for (int off = 16; off >= 1; off >>= 1)
        lsum += __shfl_xor(lsum, off, 32);
    if (lane == 0) wsum[wave] = lsum;
    __syncthreads();
    if (tid == 0) {
        float s = 0.0f;
#pragma unroll
        for (int w = 0; w < WPB; ++w) s += wsum[w];
        partials[blockIdx.x] = s;
    }
}

__global__ void finalize_kernel(const float* __restrict__ partials,
                                float* __restrict__ out) {
    // serial, deterministic sum of 512 block partials
    float s = 0.0f;
    for (int i = 0; i < GRID; ++i) s += partials[i];
    out[0] = -s * (1.0f / (float)BATCH);
}

extern "C" void kernel_launch(void* const* d_in, const int* in_sizes, int n_in,
                              void* d_out, int out_size, void* d_ws, size_t ws_size,
                              hipStream_t stream) {
    const float* W_in  = (const float*)d_in[0];
    const float* W_ctx = (const float*)d_in[1];
    const int*   iw    = (const int*)d_in[2];
    const int*   cw    = (const int*)d_in[3];
    const int*   ne    = (const int*)d_in[4];
    float* partials = (float*)d_ws;          // GRID floats = 2 KB scratch
    float* out      = (float*)d_out;

    hipLaunchKernelGGL(skipgram_wmma_kernel, dim3(GRID), dim3(BLOCK), 0, stream,
                       W_in, W_ctx, iw, cw, ne, partials);
    hipLaunchKernelGGL(finalize_kernel, dim3(1), dim3(1), 0, stream, partials, out);
}